// CosFace_77790447665629
// MI455X (gfx1250) — compile-verified
//
#include <hip/hip_runtime.h>
#include <hip/hip_bf16.h>
#include <math.h>

// ---------------------------------------------------------------------------
// CosFace loss, fused single-pass implementation for gfx1250 (MI455X).
//   cos   = norm(E) @ norm(W)^T           (fp32 WMMA 16x16x4)
//   logit = 30 * (clip(cos) - 0.4*onehot(gt))
//   loss  = mean( logsumexp(logit) - logit_gt )
// logits < 30 always => fixed-shift logsumexp = 30 + log(sum exp(logit-30)):
// single pass, per-row sums accumulate in registers.
// Weight tiles stream through LDS via GLOBAL_LOAD_ASYNC_TO_LDS_B128 with a
// triple-buffered half-K pipeline tracked by ASYNCcnt.
// LDS layout: B chunk buffers FIRST (small addresses -> DS immediate offsets
// in the hot loop), A block after (large base folded into per-lane pointer).
// ---------------------------------------------------------------------------

#define DDIM        512          // feature dim (fixed by reference)
#define AS          516          // A LDS row stride (pad 4 -> conflict-free b64)
#define BS          260          // B chunk row stride (256 K + pad 4)
#define KCH         256          // K per pipeline chunk (2 chunks per tile)
#define NBUF        3            // chunk buffers
#define ROWS_PER_WG 128
#define SPLITS      32
#define SCALE_F     30.0f
#define MARGIN_F    0.4f
#define EPS_F       1e-7f

typedef __attribute__((ext_vector_type(2))) float v2f;
typedef __attribute__((ext_vector_type(8))) float v8f;
typedef __attribute__((ext_vector_type(4))) int   v4i;

#if __has_builtin(__builtin_amdgcn_global_load_async_to_lds_b128)
#define HAVE_ASYNC 1
#else
#define HAVE_ASYNC 0
#endif

static __device__ __forceinline__ void async16(const float* g, float* l) {
#if HAVE_ASYNC
    __builtin_amdgcn_global_load_async_to_lds_b128(
        (v4i __attribute__((address_space(1)))*)(g),
        (v4i __attribute__((address_space(3)))*)(l), 0, 0);
#else
    *(float4*)l = *(const float4*)g;                 // sync fallback
#endif
}

static __device__ __forceinline__ void wait_async_le4() {
#if HAVE_ASYNC
#if __has_builtin(__builtin_amdgcn_s_wait_asynccnt)
    __builtin_amdgcn_s_wait_asynccnt(4);
#else
    asm volatile("s_wait_asynccnt 0x4" ::: "memory");
#endif
#endif
}
static __device__ __forceinline__ void wait_async_le0() {
#if HAVE_ASYNC
#if __has_builtin(__builtin_amdgcn_s_wait_asynccnt)
    __builtin_amdgcn_s_wait_asynccnt(0);
#else
    asm volatile("s_wait_asynccnt 0x0" ::: "memory");
#endif
#endif
}

static __device__ __forceinline__ v8f wmma_f32_k4(v2f a, v2f b, v8f c) {
    return __builtin_amdgcn_wmma_f32_16x16x4_f32(false, a, false, b,
                                                 (short)0, c, false, false);
}

// --- 1/||row||_2 for a [rows, 512] f32 matrix; one wave per row -------------
__global__ void rownorm_kernel(const float* __restrict__ mat,
                               float* __restrict__ inv, int rows) {
    const int wave = threadIdx.x >> 5;
    const int lane = threadIdx.x & 31;
    const int row  = blockIdx.x * 8 + wave;
    if (row >= rows) return;
    const float4* p = (const float4*)(mat + (size_t)row * DDIM);
    float ss = 0.0f;
    #pragma unroll
    for (int i = lane; i < DDIM / 4; i += 32) {
        float4 v = p[i];
        ss += v.x * v.x + v.y * v.y + v.z * v.z + v.w * v.w;
    }
    #pragma unroll
    for (int m = 16; m >= 1; m >>= 1) ss += __shfl_xor(ss, m, 32);
    if (lane == 0) inv[row] = 1.0f / fmaxf(sqrtf(ss), 1e-12f);
}

// --- main fused GEMM + margin + partial softmax -----------------------------
// grid.x = SPLITS * nRowBlocks; consecutive blocks share a class range (L2).
__global__ __launch_bounds__(256, 1)
void cosface_main(const float* __restrict__ emb, const float* __restrict__ wgt,
                  const int* __restrict__ gt, const float* __restrict__ invE,
                  const float* __restrict__ invW, float* __restrict__ rowsum,
                  float* __restrict__ gtlogit, int C, int tilesPerSplit,
                  int numTiles, int nRowBlocks) {
    extern __shared__ float lds[];
    float* Blds = lds;                          // NBUF x [16][BS] chunk buffers
    float* Alds = lds + NBUF * 16 * BS;         // [128][AS]

    const int split   = blockIdx.x / nRowBlocks;
    const int rb      = blockIdx.x % nRowBlocks;
    const int rowBase = rb * ROWS_PER_WG;
    const int tid     = threadIdx.x;
    const int wave    = tid >> 5;
    const int lane    = tid & 31;
    const int hi      = lane >> 4;            // half-wave (K split / M+8 split)
    const int col     = lane & 15;            // N index (A/B row within frags)

    const int tStart = split * tilesPerSplit;
    const int tEnd   = min(tStart + tilesPerSplit, numTiles);
    const int nCh    = 2 * (tEnd - tStart);   // half-K chunks to process

    // Stage A block (128 rows x 512) via async copies (unscaled).
    for (int i = tid; i < ROWS_PER_WG * (DDIM / 4); i += 256) {
        const int r  = i >> 7;
        const int c4 = i & 127;
        async16(emb + (size_t)(rowBase + r) * DDIM + c4 * 4,
                Alds + r * AS + c4 * 4);
    }

    // Issue a half-K chunk of the B tile: chunk ci -> buffer ci % NBUF.
    auto issue_chunk = [&](int ci) {
        const int tile = tStart + (ci >> 1);
        const int half = ci & 1;
        float* buf = Blds + (ci % NBUF) * (16 * BS);
        const int cBase = tile * 16;
        #pragma unroll
        for (int i = tid; i < 16 * (KCH / 4); i += 256) {   // 4 per thread
            const int r  = i >> 6;
            const int c4 = i & 63;
            int cls = cBase + r; if (cls >= C) cls = C - 1; // clamp, mask later
            async16(wgt + (size_t)cls * DDIM + half * KCH + c4 * 4,
                    buf + r * BS + c4 * 4);
        }
    };

    // Per-wave output geometry (16x16 f32 C/D: VGPR v -> M = v + 8*hi).
    const int myRow0 = rowBase + wave * 16 + 8 * hi;
    int   gts[8];
    float invEr[8];
    #pragma unroll
    for (int v = 0; v < 8; ++v) { gts[v] = gt[myRow0 + v]; invEr[v] = invE[myRow0 + v]; }
    float sums[8];
    #pragma unroll
    for (int v = 0; v < 8; ++v) sums[v] = 0.0f;

    // A frag: M = col, K pair at 2*hi.   B frag: N = col, K pair at 2*hi.
    const float* ap = Alds + (wave * 16 + col) * AS + 2 * hi;

    issue_chunk(0);
    if (nCh > 1) issue_chunk(1);

    v8f a0 = {0.f,0.f,0.f,0.f,0.f,0.f,0.f,0.f};
    v8f a1 = a0, a2 = a0, a3 = a0;            // 4 chains break wmma RAW dep

    for (int ci = 0; ci < nCh; ++ci) {
        if (ci + 1 < nCh) wait_async_le4();   // chunk ci landed; ci+1 may fly
        else              wait_async_le0();
        __syncthreads();                      // visibility + WAR on buffers
        if (ci + 2 < nCh) issue_chunk(ci + 2);

        const float* bp  = Blds + (ci % NBUF) * (16 * BS) + col * BS + 2 * hi;
        const float* ap2 = ap + (ci & 1) * KCH;
        #pragma unroll 4
        for (int k = 0; k < KCH; k += 16) {
            a0 = wmma_f32_k4(*(const v2f*)(ap2 + k +  0), *(const v2f*)(bp + k +  0), a0);
            a1 = wmma_f32_k4(*(const v2f*)(ap2 + k +  4), *(const v2f*)(bp + k +  4), a1);
            a2 = wmma_f32_k4(*(const v2f*)(ap2 + k +  8), *(const v2f*)(bp + k +  8), a2);
            a3 = wmma_f32_k4(*(const v2f*)(ap2 + k + 12), *(const v2f*)(bp + k + 12), a3);
        }

        if (ci & 1) {                         // tile complete: epilogue
            const v8f acc = (a0 + a1) + (a2 + a3);
            const int  cBase = (tStart + (ci >> 1)) * 16;
            const int  cls   = cBase + col;
            const bool valid = (cls < C);
            const float iw   = invW[valid ? cls : (C - 1)];
            #pragma unroll
            for (int v = 0; v < 8; ++v) {
                float cosv = acc[v] * (invEr[v] * iw);
                cosv = fminf(fmaxf(cosv, -1.0f + EPS_F), 1.0f - EPS_F);
                float logit = SCALE_F * cosv;
                if (valid && cls == gts[v]) {
                    logit -= SCALE_F * MARGIN_F;          // margin at GT
                    gtlogit[myRow0 + v] = logit;          // unique writer
                }
                sums[v] += valid ? expf(logit - SCALE_F) : 0.0f;
            }
            a0 = a1 = a2 = a3 = v8f{0.f,0.f,0.f,0.f,0.f,0.f,0.f,0.f};
        }
    }

    // Reduce sums over the 16 lanes of each half-wave, one atomic per row.
    #pragma unroll
    for (int v = 0; v < 8; ++v) {
        float s = sums[v];
        s += __shfl_xor(s, 1, 32);
        s += __shfl_xor(s, 2, 32);
        s += __shfl_xor(s, 4, 32);
        s += __shfl_xor(s, 8, 32);
        if (col == 0) atomicAdd(&rowsum[myRow0 + v], s);
    }
}

// --- final mean NLL ---------------------------------------------------------
__global__ void finalize_kernel(const float* __restrict__ rowsum,
                                const float* __restrict__ gtlogit,
                                float* __restrict__ out, int N) {
    __shared__ float red[256];
    float s = 0.0f;
    for (int r = threadIdx.x; r < N; r += 256)
        s += SCALE_F + logf(rowsum[r]) - gtlogit[r];
    red[threadIdx.x] = s;
    __syncthreads();
    for (int off = 128; off > 0; off >>= 1) {
        if (threadIdx.x < off) red[threadIdx.x] += red[threadIdx.x + off];
        __syncthreads();
    }
    if (threadIdx.x == 0) out[0] = red[0] / (float)N;
}

extern "C" void kernel_launch(void* const* d_in, const int* in_sizes, int n_in,
                              void* d_out, int out_size, void* d_ws, size_t ws_size,
                              hipStream_t stream) {
    const float* emb = (const float*)d_in[0];
    const int*   gt  = (const int*)d_in[1];
    const float* wgt = (const float*)d_in[2];
    float*       out = (float*)d_out;

    const int N = in_sizes[1];            // 2048
    const int D = in_sizes[0] / N;        // 512 (DDIM)
    const int C = in_sizes[2] / D;        // 100000
    (void)n_in; (void)out_size; (void)ws_size;

    float* ws      = (float*)d_ws;
    float* rowsum  = ws;                  // [N]
    float* gtlog   = ws + N;              // [N]
    float* invE    = ws + 2 * N;          // [N]
    float* invW    = ws + 3 * N;          // [C]

    (void)hipMemsetAsync(rowsum, 0, (size_t)N * sizeof(float), stream);

    rownorm_kernel<<<(N + 7) / 8, 256, 0, stream>>>(emb, invE, N);
    rownorm_kernel<<<(C + 7) / 8, 256, 0, stream>>>(wgt, invW, C);

    const int numTiles = (C + 15) / 16;
    const int nRB      = N / ROWS_PER_WG;                 // 16
    const int tps      = (numTiles + SPLITS - 1) / SPLITS;
    const size_t ldsBytes =
        (size_t)(ROWS_PER_WG * AS + NBUF * 16 * BS) * sizeof(float); // 314,112B

    (void)hipFuncSetAttribute((const void*)cosface_main,
                              hipFuncAttributeMaxDynamicSharedMemorySize,
                              (int)ldsBytes);
    cosface_main<<<SPLITS * nRB, 256, ldsBytes, stream>>>(
        emb, wgt, gt, invE, invW, rowsum, gtlog, C, tps, numTiles, nRB);

    finalize_kernel<<<1, 256, 0, stream>>>(rowsum, gtlog, out, N);
}